// GMELoss3D_45148696216030
// MI455X (gfx1250) — compile-verified
//
#include <hip/hip_runtime.h>
#include <math.h>

typedef __attribute__((ext_vector_type(2))) float v2f;
typedef __attribute__((ext_vector_type(8))) float v8f;

#define DIM 128
#define TX 8
#define TY 8
#define TZ 16
#define ZP 18            // LDS innermost extent (16 + 2 halo)
#define LDSZ (10*10*ZP)  // (8+2) x (8+2) x 18 haloed tile

// Padded WMMA A-matrix weight table: 16 rows (stencils 0..8, rows 9..15 zero),
// 28 columns (taps t=(d0*3+d1)*3+d2 for d* in {0,1,2}; column 27 = K-pad, zero).
// Stencils: Sx, Sy, Sz, Sd11, Sd12, Sd21=Sd11.T, Sd22=Sd12.T,
//           Sd31[a][b][c]=-Sd11[b][c][a], Sd32[a][b][c]=Sd12[b][c][a].
// n1=1, n2=2, n3=2 (n3*n2=4).
__constant__ float WTBL[16 * 28] = {
  // m=0: Sx
  -1,0,1, -2,0,2, -1,0,1,   -2,0,2, -4,0,4, -2,0,2,   -1,0,1, -2,0,2, -1,0,1,  0,
  // m=1: Sy
  -1,-2,-1, 0,0,0, 1,2,1,   -2,-4,-2, 0,0,0, 2,4,2,   -1,-2,-1, 0,0,0, 1,2,1,  0,
  // m=2: Sz
  -1,-2,-1, -2,-4,-2, -1,-2,-1,  0,0,0, 0,0,0, 0,0,0,  1,2,1, 2,4,2, 1,2,1,    0,
  // m=3: Sd11
  0,1,2, -1,0,1, -2,-1,0,   0,2,4, -2,0,2, -4,-2,0,   0,1,2, -1,0,1, -2,-1,0,  0,
  // m=4: Sd12
  -2,-1,0, -1,0,1, 0,1,2,   -4,-2,0, -2,0,2, 0,2,4,   -2,-1,0, -1,0,1, 0,1,2,  0,
  // m=5: Sd21 = Sd11.T
  0,0,0, -1,-2,-1, -2,-4,-2,   1,2,1, 0,0,0, -1,-2,-1,   2,4,2, 1,2,1, 0,0,0,  0,
  // m=6: Sd22 = Sd12.T
  -2,-4,-2, -1,-2,-1, 0,0,0,   -1,-2,-1, 0,0,0, 1,2,1,   0,0,0, 1,2,1, 2,4,2,  0,
  // m=7: Sd31
  0,1,2, 0,2,4, 0,1,2,   -1,0,1, -2,0,2, -1,0,1,   -2,-1,0, -4,-2,0, -2,-1,0,  0,
  // m=8: Sd32
  -2,-1,0, -4,-2,0, -2,-1,0,   -1,0,1, -2,0,2, -1,0,1,   0,1,2, 0,2,4, 0,1,2,  0,
  // m=9..15: zero rows (pad M to 16)
  0,0,0,0,0,0,0,0,0, 0,0,0,0,0,0,0,0,0, 0,0,0,0,0,0,0,0,0, 0,
  0,0,0,0,0,0,0,0,0, 0,0,0,0,0,0,0,0,0, 0,0,0,0,0,0,0,0,0, 0,
  0,0,0,0,0,0,0,0,0, 0,0,0,0,0,0,0,0,0, 0,0,0,0,0,0,0,0,0, 0,
  0,0,0,0,0,0,0,0,0, 0,0,0,0,0,0,0,0,0, 0,0,0,0,0,0,0,0,0, 0,
  0,0,0,0,0,0,0,0,0, 0,0,0,0,0,0,0,0,0, 0,0,0,0,0,0,0,0,0, 0,
  0,0,0,0,0,0,0,0,0, 0,0,0,0,0,0,0,0,0, 0,0,0,0,0,0,0,0,0, 0,
  0,0,0,0,0,0,0,0,0, 0,0,0,0,0,0,0,0,0, 0,0,0,0,0,0,0,0,0, 0
};

__global__ __launch_bounds__(256)
void gme_main(const float* __restrict__ Y, const float* __restrict__ P,
              float* __restrict__ partials) {
  __shared__ float ty[LDSZ];
  __shared__ float tp[LDSZ];
  __shared__ float red[8];

  const int tid  = threadIdx.x;
  const int lane = tid & 31;
  const int wave = tid >> 5;
  const int n    = lane & 15;        // voxel index (B/N) == stencil row (A/M)

  const int zt = blockIdx.x, yt = blockIdx.y;
  const int xt = blockIdx.z & 15, b = blockIdx.z >> 4;
  const int x0 = xt * TX, y0 = yt * TY, z0 = zt * TZ;

  const size_t chan = (size_t)DIM * DIM * DIM;
  const float* yb = Y + (size_t)b * 2u * chan;
  const float* pb = P + (size_t)b * 2u * chan;

  // Cooperative haloed load; sum the 2 channels; zero-pad at volume edges (SAME).
  for (int idx = tid; idx < LDSZ; idx += 256) {
    int xi = idx / (10 * ZP);
    int r  = idx - xi * (10 * ZP);
    int yi = r / ZP;
    int zi = r - yi * ZP;
    int g0 = x0 + xi - 1, g1 = y0 + yi - 1, g2 = z0 + zi - 1;
    float vy = 0.f, vp = 0.f;
    if ((unsigned)g0 < (unsigned)DIM && (unsigned)g1 < (unsigned)DIM &&
        (unsigned)g2 < (unsigned)DIM) {
      size_t o = ((size_t)g0 * DIM + g1) * DIM + g2;
      vy = yb[o] + yb[o + chan];
      vp = pb[o] + pb[o + chan];
    }
    ty[idx] = vy;
    tp[idx] = vp;
  }
  __syncthreads();

  // Per-lane constants for the 7 K=4 chunks covering taps 0..26 (tap 27 padded).
  // A layout (16x4 f32): VGPR0 = K0 (lanes<16) / K2 (lanes>=16); VGPR1 = K1/K3.
  // B assumed mirrored. Weight table is zero-padded, so no guards needed.
  const int kbase = (lane >= 16) ? 2 : 0;
  float aw0[7], aw1[7];
  int   of0[7], of1[7];
#pragma unroll
  for (int c = 0; c < 7; ++c) {
    int t0 = 4 * c + kbase, t1 = t0 + 1;
    aw0[c] = WTBL[n * 28 + t0];
    aw1[c] = WTBL[n * 28 + t1];
    int u0 = t0 > 26 ? 26 : t0;            // clamp pad tap to a valid LDS address
    int u1 = t1 > 26 ? 26 : t1;
    {
      int dz = u0 % 3, dy = (u0 / 3) % 3, dx = u0 / 9;
      of0[c] = (dx * 10 + dy) * ZP + dz + n;
    }
    {
      int dz = u1 % 3, dy = (u1 / 3) % 3, dx = u1 / 9;
      of1[c] = (dx * 10 + dy) * ZP + dz + n;
    }
  }

  float lsum = 0.f;
  // Each wave owns x-row `wave`; iterates the 8 y-lines of 16 z-voxels.
#pragma unroll 1
  for (int i = 0; i < 8; ++i) {
    const int base = (wave * 10 + i) * ZP;   // LDS index of (x=wave,y=i,z=0) w/ halo origin
    v8f accY = {0,0,0,0,0,0,0,0};
    v8f accP = {0,0,0,0,0,0,0,0};
#pragma unroll
    for (int c = 0; c < 7; ++c) {
      v2f a; a.x = aw0[c]; a.y = aw1[c];
      v2f bv; bv.x = ty[base + of0[c]]; bv.y = ty[base + of1[c]];
      accY = __builtin_amdgcn_wmma_f32_16x16x4_f32(false, a, false, bv,
                                                   (short)0, accY, false, false);
    }
#pragma unroll
    for (int c = 0; c < 7; ++c) {
      v2f a; a.x = aw0[c]; a.y = aw1[c];
      v2f bv; bv.x = tp[base + of0[c]]; bv.y = tp[base + of1[c]];
      accP = __builtin_amdgcn_wmma_f32_16x16x4_f32(false, a, false, bv,
                                                   (short)0, accP, false, false);
    }
    // D layout: VGPR j = stencil row j (lanes<16) / j+8 (lanes>=16). Rows 9..15 are 0.
    float sY = 0.f, sP = 0.f;
#pragma unroll
    for (int j = 0; j < 8; ++j) { sY += accY[j] * accY[j]; sP += accP[j] * accP[j]; }
    sY += __shfl_xor(sY, 16, 32);
    sP += __shfl_xor(sP, 16, 32);
    if (lane < 16) {
      float mY = sqrtf(sY + 1e-12f) * 0.5f;   // /C with C=2
      float mP = sqrtf(sP + 1e-12f) * 0.5f;
      float d = mY - mP;
      lsum += d * d;
    }
  }

  // Deterministic block reduction -> one partial per block.
#pragma unroll
  for (int o = 16; o > 0; o >>= 1) lsum += __shfl_down(lsum, o, 32);
  if (lane == 0) red[wave] = lsum;
  __syncthreads();
  if (tid == 0) {
    float t = 0.f;
#pragma unroll
    for (int w = 0; w < 8; ++w) t += red[w];
    int blin = (blockIdx.z * gridDim.y + blockIdx.y) * gridDim.x + blockIdx.x;
    partials[blin] = t;
  }
}

__global__ __launch_bounds__(256)
void gme_reduce(const float* __restrict__ partials, float* __restrict__ out,
                int n, float scale) {
  __shared__ float s[256];
  float t = 0.f;
  for (int i = threadIdx.x; i < n; i += 256) t += partials[i];
  s[threadIdx.x] = t;
  __syncthreads();
  for (int o = 128; o > 0; o >>= 1) {
    if (threadIdx.x < o) s[threadIdx.x] += s[threadIdx.x + o];
    __syncthreads();
  }
  if (threadIdx.x == 0) out[0] = s[0] * scale;
}

extern "C" void kernel_launch(void* const* d_in, const int* in_sizes, int n_in,
                              void* d_out, int out_size, void* d_ws, size_t ws_size,
                              hipStream_t stream) {
  const float* y  = (const float*)d_in[0];
  const float* yp = (const float*)d_in[1];
  float* out      = (float*)d_out;
  float* partials = (float*)d_ws;    // 4096 floats = 16 KB of scratch

  dim3 grid(DIM / TZ, DIM / TY, (DIM / TX) * 2);   // (8, 16, 32) = 4096 blocks
  gme_main<<<grid, 256, 0, stream>>>(y, yp, partials);

  const int nblocks = (DIM / TZ) * (DIM / TY) * (DIM / TX) * 2;
  const float scale = 1.0f / (2.0f * DIM * DIM * DIM);  // mean over B*1*H*W*D
  gme_reduce<<<1, 256, 0, stream>>>(partials, out, nblocks, scale);
}